// CIFModule_11304353923362
// MI455X (gfx1250) — compile-verified
//
#include <hip/hip_runtime.h>

// ---------------- problem constants ----------------
#define Bq    16
#define Tq    3000
#define Dq    512
#define NTOK  448
#define LN_EPS 1e-5f

// alpha kernel tiling: 24 timesteps + 2 halo rows in LDS (26*512*4 = 53 KB)
#define T_TILE   24
#define AROWS    (T_TILE + 2)
#define NTILES_A ((Tq + T_TILE - 1) / T_TILE)   // 125 (exact)

// gather kernel tiling: 24 timesteps staged per chunk (24*512*4 = 48 KB)
#define TCH   24
#define NGRP  (NTOK / 16)                       // 28 bin-groups per batch

typedef __attribute__((ext_vector_type(2))) float v2f;
typedef __attribute__((ext_vector_type(8))) float v8f;

// =====================================================================
// Kernel 1: depthwise conv (K=3, pad 1) + LayerNorm + proj + sigmoid
//   logit = rsqrt(var+eps) * (S_ygw - mu*S_gw) + S_bw + proj_b
// One block = 24 timesteps of one batch; 8 waves x 3 timesteps each.
// Each lane owns 16 channels; butterfly wave32 reductions.
// =====================================================================
__global__ __launch_bounds__(256) void alpha_kernel(
    const float* __restrict__ enc, const float* __restrict__ conv_w,
    const float* __restrict__ gamma, const float* __restrict__ beta,
    const float* __restrict__ proj_w, const float* __restrict__ proj_b,
    const int* __restrict__ in_len, float* __restrict__ alpha_out)
{
    __shared__ __align__(16) float sb[AROWS * Dq];
    const int b    = blockIdx.x / NTILES_A;
    const int tile = blockIdx.x % NTILES_A;
    const int t0   = tile * T_TILE;
    const int tid  = threadIdx.x;

    const float* encb = enc + (size_t)b * Tq * Dq;
    // stage rows [t0-1 .. t0+T_TILE] (zero-padded at tensor edges)
    for (int idx = tid; idx < AROWS * (Dq / 4); idx += 256) {
        int row = idx / (Dq / 4);
        int c4  = idx % (Dq / 4);
        int gt  = t0 - 1 + row;
        float4 v = make_float4(0.f, 0.f, 0.f, 0.f);
        if (gt >= 0 && gt < Tq)
            v = *reinterpret_cast<const float4*>(encb + (size_t)gt * Dq + c4 * 4);
        *reinterpret_cast<float4*>(&sb[row * Dq + c4 * 4]) = v;
    }
    __syncthreads();

    const int wave  = tid >> 5;
    const int lane  = tid & 31;
    const int cbase = lane * 16;

    float w0[16], w1[16], w2[16], gw[16];
    float s_gw = 0.f, s_bw = 0.f;
    for (int i = 0; i < 16; ++i) {
        int c = cbase + i;
        w0[i] = conv_w[c * 3 + 0];
        w1[i] = conv_w[c * 3 + 1];
        w2[i] = conv_w[c * 3 + 2];
        float g = gamma[c], pw = proj_w[c];
        gw[i] = g * pw;
        s_gw += gw[i];
        s_bw += beta[c] * pw;
    }
    for (int m = 16; m; m >>= 1) {
        s_gw += __shfl_xor(s_gw, m, 32);
        s_bw += __shfl_xor(s_bw, m, 32);
    }
    const float pb  = proj_b[0];
    const int   len = in_len[b];

    for (int i = 0; i < 3; ++i) {
        int lt = wave * 3 + i;
        int t  = t0 + lt;
        if (t >= Tq) break;
        const float* r0 = &sb[lt * Dq + cbase];   // t-1
        const float* r1 = r0 + Dq;                // t
        const float* r2 = r1 + Dq;                // t+1
        float s1 = 0.f, s2 = 0.f, s3 = 0.f;
        for (int k = 0; k < 16; ++k) {
            float y = w0[k] * r0[k] + w1[k] * r1[k] + w2[k] * r2[k];
            s1 += y;
            s2 += y * y;
            s3 += y * gw[k];
        }
        for (int m = 16; m; m >>= 1) {
            s1 += __shfl_xor(s1, m, 32);
            s2 += __shfl_xor(s2, m, 32);
            s3 += __shfl_xor(s3, m, 32);
        }
        if (lane == 0) {
            float mu    = s1 * (1.f / Dq);
            float var   = s2 * (1.f / Dq) - mu * mu;
            float rs    = rsqrtf(var + LN_EPS);
            float logit = rs * (s3 - mu * s_gw) + s_bw + pb;
            float a     = (t < len) ? 1.f / (1.f + expf(-logit)) : 0.f;
            alpha_out[(size_t)b * Tq + t] = a;
        }
    }
}

// =====================================================================
// Kernel 2: per-batch sum(alpha), rescale factor, quantity loss.
// One block (16 waves); wave b reduces batch b.
// =====================================================================
__global__ __launch_bounds__(512) void stats_kernel(
    const float* __restrict__ alpha, const int* __restrict__ tgt_len,
    float* __restrict__ ws_scale, float* __restrict__ ws_diff,
    float* __restrict__ qty_out)
{
    int tid = threadIdx.x, b = tid >> 5, lane = tid & 31;
    float s = 0.f;
    for (int t = lane; t < Tq; t += 32) s += alpha[(size_t)b * Tq + t];
    for (int m = 16; m; m >>= 1) s += __shfl_xor(s, m, 32);
    if (lane == 0) {
        float tgt = (float)tgt_len[b];
        ws_scale[b] = tgt / fmaxf(s, 1e-8f);
        ws_diff[b]  = fabsf(s - tgt);
    }
    __syncthreads();
    if (tid == 0) {
        float q = 0.f;
        for (int i = 0; i < Bq; ++i) q += ws_diff[i];
        qty_out[0] = q * (1.f / Bq);
    }
}

// =====================================================================
// Kernel 3: inclusive cumsum of alpha*scale along T. One wave per batch,
// chunked wave scan with carried prefix.
// =====================================================================
__global__ __launch_bounds__(32) void cumsum_kernel(
    const float* __restrict__ alpha, const float* __restrict__ ws_scale,
    float* __restrict__ ws_cum)
{
    int b = blockIdx.x, lane = threadIdx.x;
    float scale = ws_scale[b];
    float carry = 0.f;
    for (int t0 = 0; t0 < Tq; t0 += 32) {
        int t = t0 + lane;
        float v = (t < Tq) ? alpha[(size_t)b * Tq + t] * scale : 0.f;
        for (int d = 1; d < 32; d <<= 1) {
            float n = __shfl_up(v, d, 32);
            if (lane >= d) v += n;
        }
        if (t < Tq) ws_cum[(size_t)b * Tq + t] = v + carry;
        carry += __shfl(v, 31, 32);
    }
}

// =====================================================================
// Kernel 4: CIF integration as a bin-centric gather-GEMM (no atomics).
// Block = (batch b, bins n0..n0+15). Monotone cum -> contiguous t-range:
//   { t : cum[t] >= n0  &&  cum[t-1] < n0+16 }   (binary searched)
// out[n0+m, :] = sum_t W[m,t] * enc[t, :]   with
//   W[m,t] = [floor(cp)==n]*max(min(c,floor(cp)+1)-cp,0)
//          + [floor(c)==n] *max(c-floor(cp)-1,0)
// Computed as f32 WMMA 16x16x4: M=16 bins, N=16 channels, K=timesteps.
// 8 waves x 64 channels cover D=512; each wave holds 4 v8f accumulators.
// =====================================================================
__global__ __launch_bounds__(256) void gather_kernel(
    const float* __restrict__ enc, const float* __restrict__ ws_cum,
    float* __restrict__ out)
{
    __shared__ __align__(16) float ebuf[TCH * Dq];   // 48 KB encoder stage
    __shared__ float wbuf[16 * TCH];                 // 16 bins x 24 timesteps

    const int b   = blockIdx.x / NGRP;
    const int n0  = (blockIdx.x % NGRP) * 16;
    const int tid = threadIdx.x;
    const float* cum  = ws_cum + (size_t)b * Tq;
    const float* encb = enc + (size_t)b * Tq * Dq;

    const float lo_v = (float)n0;
    const float hi_v = (float)(n0 + 16);
    int tA, tB;
    {   // first t with cum[t] >= n0
        int l = 0, h = Tq;
        while (l < h) { int m = (l + h) >> 1; if (cum[m] >= lo_v) h = m; else l = m + 1; }
        tA = l;
    }
    {   // first t with cum[t-1] >= n0+16  (cum[-1] == 0)
        int l = 0, h = Tq;
        while (l < h) { int m = (l + h) >> 1; if (cum[m] >= hi_v) h = m; else l = m + 1; }
        tB = (l + 1 < Tq) ? (l + 1) : Tq;
    }

    const int wave = tid >> 5;
    const int lane = tid & 31;
    const int hw   = lane >> 4;        // half-wave select
    const int ln16 = lane & 15;
    const int kq   = hw * 2;           // f32 WMMA K slots: {0,1} / {2,3}
    const int chb  = wave * 64 + ln16; // channel base for this lane

    v8f acc0 = {}, acc1 = {}, acc2 = {}, acc3 = {};

    for (int tc = tA; tc < tB; tc += TCH) {
        int nt = tB - tc; if (nt > TCH) nt = TCH;   // block-uniform
        __syncthreads();
        // stage encoder rows (coalesced float4), zero-fill past range
        for (int idx = tid; idx < TCH * (Dq / 4); idx += 256) {
            int row = idx / (Dq / 4);
            int c4  = idx % (Dq / 4);
            float4 v = make_float4(0.f, 0.f, 0.f, 0.f);
            if (row < nt)
                v = *reinterpret_cast<const float4*>(encb + (size_t)(tc + row) * Dq + c4 * 4);
            *reinterpret_cast<float4*>(&ebuf[row * Dq + c4 * 4]) = v;
        }
        // build weight matrix W[16][TCH]
        for (int idx = tid; idx < 16 * TCH; idx += 256) {
            int m = idx / TCH, j = idx % TCH;
            float w = 0.f;
            int t = tc + j;
            if (j < nt) {
                float cp  = (t == 0) ? 0.f : cum[t - 1];
                float cc  = cum[t];
                int   n   = n0 + m;
                int   klo = (int)floorf(cp);
                if (klo == n)
                    w += fmaxf(fminf(cc, (float)(klo + 1)) - cp, 0.f);
                if ((int)floorf(cc) == n)
                    w += fmaxf(cc - (float)(klo + 1), 0.f);
            }
            wbuf[m * TCH + j] = w;
        }
        if (tc + TCH < tB)  // prefetch next chunk -> global_prefetch_b8
            __builtin_prefetch(encb + (size_t)(tc + TCH) * Dq + (tid << 2), 0, 1);
        __syncthreads();

        for (int j4 = 0; j4 < TCH / 4; ++j4) {
            int k0 = j4 * 4;
            if (k0 >= nt) break;                   // block-uniform break
            // A: W tile, f32 16x4 layout (lane l%16 = row, K = (l/16)*2+e)
            v2f a;
            a.x = wbuf[ln16 * TCH + k0 + kq];
            a.y = wbuf[ln16 * TCH + k0 + kq + 1];
            // B: encoder tile 4x16 (row K = (l/16)*2+e, col = l%16)
            const float* e0 = &ebuf[(k0 + kq) * Dq + chb];
            const float* e1 = e0 + Dq;
            v2f b0, b1, b2, b3;
            b0.x = e0[0];  b0.y = e1[0];
            b1.x = e0[16]; b1.y = e1[16];
            b2.x = e0[32]; b2.y = e1[32];
            b3.x = e0[48]; b3.y = e1[48];
            acc0 = __builtin_amdgcn_wmma_f32_16x16x4_f32(false, a, false, b0, (short)0, acc0, false, false);
            acc1 = __builtin_amdgcn_wmma_f32_16x16x4_f32(false, a, false, b1, (short)0, acc1, false, false);
            acc2 = __builtin_amdgcn_wmma_f32_16x16x4_f32(false, a, false, b2, (short)0, acc2, false, false);
            acc3 = __builtin_amdgcn_wmma_f32_16x16x4_f32(false, a, false, b3, (short)0, acc3, false, false);
        }
    }

    // D layout: VGPR j -> row j (lanes 0-15) / row j+8 (lanes 16-31), col = lane%16
    const int mbase = n0 + hw * 8;
    float* ob = out + (size_t)b * NTOK * Dq;
    for (int j = 0; j < 8; ++j) {
        size_t r = (size_t)(mbase + j) * Dq + wave * 64 + ln16;
        ob[r +  0] = acc0[j];
        ob[r + 16] = acc1[j];
        ob[r + 32] = acc2[j];
        ob[r + 48] = acc3[j];
    }
}

// =====================================================================
extern "C" void kernel_launch(void* const* d_in, const int* in_sizes, int n_in,
                              void* d_out, int out_size, void* d_ws, size_t ws_size,
                              hipStream_t stream) {
    const float* enc     = (const float*)d_in[0];
    const float* conv_w  = (const float*)d_in[1];
    const float* gamma   = (const float*)d_in[2];
    const float* beta    = (const float*)d_in[3];
    const float* proj_w  = (const float*)d_in[4];
    const float* proj_b  = (const float*)d_in[5];
    const int*   in_len  = (const int*)d_in[6];
    const int*   tgt_len = (const int*)d_in[7];

    float* out   = (float*)d_out;                      // [B, 448, 512]
    float* alpha = out + (size_t)Bq * NTOK * Dq;       // [B, T]
    float* qty   = alpha + (size_t)Bq * Tq;            // scalar

    float* ws       = (float*)d_ws;
    float* ws_cum   = ws;                              // B*T
    float* ws_scale = ws + (size_t)Bq * Tq;            // B
    float* ws_diff  = ws_scale + Bq;                   // B

    alpha_kernel <<<Bq * NTILES_A, 256, 0, stream>>>(enc, conv_w, gamma, beta,
                                                     proj_w, proj_b, in_len, alpha);
    stats_kernel <<<1, 512, 0, stream>>>(alpha, tgt_len, ws_scale, ws_diff, qty);
    cumsum_kernel<<<Bq, 32, 0, stream>>>(alpha, ws_scale, ws_cum);
    gather_kernel<<<Bq * NGRP, 256, 0, stream>>>(enc, ws_cum, out);
}